// TemporalContrast_35287451304939
// MI455X (gfx1250) — compile-verified
//
#include <hip/hip_runtime.h>
#include <math.h>

#define GAMMA_  0.1f
#define INVG_   10.0f       // 1/GAMMA
#define ENERGY_ 0.5f
#define NSEQ    128
#define TLEN    64
#define DFEAT   13
#define KPAD    16
#define INF_    1e10f

typedef __attribute__((ext_vector_type(2))) float v2f;
typedef __attribute__((ext_vector_type(8))) float v8f;

// Stable soft-min of 3 values: -g * log(sum exp(-z/g))
__device__ __forceinline__ float softmin3(float z0, float z1, float z2) {
    float a = fminf(fminf(z0, z1), z2);
    float s = __expf((a - z0) * INVG_) + __expf((a - z1) * INVG_) + __expf((a - z2) * INVG_);
    return a - GAMMA_ * __logf(s);
}

// One block (64 threads = 2 waves) per sequence pair p = a*128 + b.
// Phase 1: stage both sequences in LDS, compute squared norms.
// Phase 2: 64x64 distance matrix via V_WMMA_F32_16X16X4_F32 (K padded 13->16).
// Phase 3: anti-diagonal soft-DTW wavefront DP (127 steps).
__global__ __launch_bounds__(64) void sdtw_pair_kernel(const float* __restrict__ xxx,
                                                       const float* __restrict__ yyy,
                                                       float* __restrict__ simR) {
    __shared__ float Afr[TLEN][KPAD + 1];   // +1 pad: conflict-free column reads
    __shared__ float Bfr[TLEN][KPAD + 1];
    __shared__ float sqA[TLEN], sqB[TLEN];
    __shared__ float Dm[TLEN * TLEN];       // stride 64: DP reads Dm[i*64 + (k-i)] -> bank (k-i)%64, distinct/lane
    __shared__ float Rm1[TLEN], Rm2[TLEN];

    const int tid  = threadIdx.x;          // 0..63
    const int lane = tid & 31;
    const int wv   = tid >> 5;             // wave 0/1
    const int p    = blockIdx.x;
    const int aIdx = p >> 7;
    const int bIdx = p & 127;

    const float* pa = (aIdx < 64) ? (xxx + aIdx * (TLEN * DFEAT))
                                  : (yyy + (aIdx - 64) * (TLEN * DFEAT));
    const float* pb = (bIdx < 64) ? (xxx + bIdx * (TLEN * DFEAT))
                                  : (yyy + (bIdx - 64) * (TLEN * DFEAT));

    // ---- Phase 1: load frames (thread t -> frame t of both sequences) ----
    {
        float sa = 0.f, sb = 0.f;
        const float* fa = pa + tid * DFEAT;
        const float* fb = pb + tid * DFEAT;
        #pragma unroll
        for (int f = 0; f < DFEAT; ++f) {
            float va = fa[f], vb = fb[f];
            Afr[tid][f] = va; Bfr[tid][f] = vb;
            sa += va * va;   sb += vb * vb;
        }
        #pragma unroll
        for (int f = DFEAT; f < KPAD; ++f) { Afr[tid][f] = 0.f; Bfr[tid][f] = 0.f; }
        sqA[tid] = sa; sqB[tid] = sb;
    }
    __syncthreads();

    // ---- Phase 2: D[i][j] = sqA[i] + sqB[j] - 2*dot(a_i, b_j) via fp32 WMMA ----
    // 4x4 grid of 16x16 tiles; wave 0 -> tiles 0..7, wave 1 -> tiles 8..15.
    // ISA fp32 A 16x4 layout: lanes 0-15 carry K=0(v0),K=1(v1); lanes 16-31 K=2,K=3.
    const int halfsel = lane >> 4;          // 0 | 1
    const int lid     = lane & 15;
    for (int t8 = 0; t8 < 8; ++t8) {
        const int tile = wv * 8 + t8;
        const int tm = (tile >> 2) * 16;    // row tile base (seq a frames)
        const int tn = (tile & 3) * 16;     // col tile base (seq b frames)
        v8f acc = {};
        #pragma unroll
        for (int kk = 0; kk < 4; ++kk) {
            const int ks = kk * 4 + halfsel * 2;
            v2f av, bv;
            av.x = Afr[tm + lid][ks];
            av.y = Afr[tm + lid][ks + 1];
            bv.x = Bfr[tn + lid][ks];
            bv.y = Bfr[tn + lid][ks + 1];
            acc = __builtin_amdgcn_wmma_f32_16x16x4_f32(
                /*neg_a=*/false, av, /*neg_b=*/false, bv,
                /*c_mod=*/(short)0, acc, /*reuse_a=*/false, /*reuse_b=*/false);
        }
        // C/D layout: VGPR r holds rows (tm + r) for lanes 0-15 and (tm + 8 + r) for lanes 16-31.
        #pragma unroll
        for (int r = 0; r < 8; ++r) {
            const int i = tm + r + halfsel * 8;
            const int j = tn + lid;
            Dm[i * TLEN + j] = sqA[i] + sqB[j] - 2.0f * acc[r];
        }
    }

    // ---- Phase 3: soft-DTW wavefront DP ----
    Rm1[tid] = INF_; Rm2[tid] = INF_;
    __syncthreads();
    const int i = tid;
    for (int k = 0; k < 2 * TLEN - 1; ++k) {
        const int j = k - i;
        float Rk = INF_;
        if (j >= 0 && j < TLEN) {
            const float r_left = Rm1[i];                                  // R[i, j-1]
            const float r_up   = (i > 0) ? Rm1[i - 1] : INF_;             // R[i-1, j]
            const float r_diag = (i > 0) ? Rm2[i - 1]
                                         : ((k == 0) ? 0.0f : INF_);      // R[i-1, j-1]
            Rk = Dm[i * TLEN + j] + softmin3(r_up, r_left, r_diag);
        }
        __syncthreads();
        Rm2[i] = Rm1[i];
        Rm1[i] = Rk;
        __syncthreads();
    }
    if (tid == TLEN - 1) simR[p] = Rm1[TLEN - 1];
}

// One block, 128 threads: row-wise logsumexp of sim (diag=0, /ENERGY),
// minus the positive term, mean over 128 rows -> scalar loss.
__global__ __launch_bounds__(128) void loss_kernel(const float* __restrict__ simR,
                                                   float* __restrict__ out) {
    __shared__ float red[NSEQ];
    const int n = threadIdx.x;              // row 0..127
    const float invE = 1.0f / ENERGY_;

    float mx = 0.0f;                        // diagonal contributes 0
    for (int m = 0; m < NSEQ; ++m) {
        const float v = (m == n) ? 0.0f : simR[n * NSEQ + m] * invE;
        mx = fmaxf(mx, v);
    }
    float s = 0.0f;
    for (int m = 0; m < NSEQ; ++m) {
        const float v = (m == n) ? 0.0f : simR[n * NSEQ + m] * invE;
        s += __expf(v - mx);
    }
    const float lse = mx + __logf(s);

    const int i0 = n & 63;                  // positives[n] = sim[i0, i0+64]
    const float positive = simR[i0 * NSEQ + (i0 + 64)] * invE;

    red[n] = lse - positive;
    __syncthreads();
    for (int off = 64; off > 0; off >>= 1) {
        if (n < off) red[n] += red[n + off];
        __syncthreads();
    }
    if (n == 0) out[0] = red[0] / (float)NSEQ;
}

extern "C" void kernel_launch(void* const* d_in, const int* in_sizes, int n_in,
                              void* d_out, int out_size, void* d_ws, size_t ws_size,
                              hipStream_t stream) {
    const float* xxx = (const float*)d_in[0];   // [64, 64, 13] f32
    const float* yyy = (const float*)d_in[1];   // [64, 64, 13] f32
    float* simR = (float*)d_ws;                 // 16384 floats (64 KB) of scratch
    float* out  = (float*)d_out;                // scalar loss

    sdtw_pair_kernel<<<dim3(NSEQ * NSEQ), dim3(64), 0, stream>>>(xxx, yyy, simR);
    loss_kernel<<<dim3(1), dim3(NSEQ), 0, stream>>>(simR, out);
}